// DistanceAttention_67018669686961
// MI455X (gfx1250) — compile-verified
//
#include <hip/hip_runtime.h>

typedef __attribute__((ext_vector_type(2))) float        v2f;
typedef __attribute__((ext_vector_type(4))) float        v4f;
typedef __attribute__((ext_vector_type(8))) float        v8f;
typedef __attribute__((ext_vector_type(4))) unsigned int v4u;
typedef __attribute__((ext_vector_type(8))) int          v8i;
typedef __attribute__((ext_vector_type(4))) int          v4i;

#define S    2048
#define RANK 256
#define NB   8
#define TM   128
#define TN   128
#define KC   64
#define LDA  68    // KC + 4 pad: 272 B rows (16B aligned), bank-conflict-free v2f frag reads
#define LDC  132   // TN + 4 pad for epilogue staging
#define BUFF (2 * TM * LDA)   // floats per double-buffer slot (A+B)

// ---------------- Kernel 1: per-row squared norms -> d_ws ----------------
__global__ __launch_bounds__(256)
void row_sqnorm_kernel(const float* __restrict__ X, float* __restrict__ nrm) {
  const int lane = threadIdx.x & 31;
  const int wid  = threadIdx.x >> 5;
  const int row  = blockIdx.x * 8 + wid;
  const float* xr = X + (size_t)row * RANK;
  float s = 0.f;
  #pragma unroll
  for (int k = lane; k < RANK; k += 32) {
    float v = xr[k];
    s = fmaf(v, v, s);
  }
  #pragma unroll
  for (int off = 16; off > 0; off >>= 1)
    s += __shfl_xor(s, off, 32);
  if (lane == 0) nrm[row] = s;
}

// ---- TDM: DMA a 128-row x 64-float tile (row stride 256 floats) into LDS ----
// D# pad fields insert 4 DWORDs after every 64 DWORDs -> LDS row stride = LDA = 68.
__device__ __forceinline__ void tdm_load_tile(const float* gsrc, unsigned lds_addr) {
  unsigned long long ga = (unsigned long long)(uintptr_t)gsrc;
  v4u g0;
  g0.x = 1u;                                        // count=1 valid descriptor
  g0.y = lds_addr;                                  // LDS byte address (bits 63:32)
  g0.z = (unsigned)ga;                              // global_addr[31:0]
  g0.w = (unsigned)(ga >> 32) | 0x80000000u;        // global_addr[56:32] | type=2 (bits 127:126)
  v8i g1;
  g1[0] = (int)0x07520000;      // data_size=2(4B) | pad_enable | pad_interval=5(64DW) | pad_amount=3(4DW)
  g1[1] = (int)(RANK << 16);    // tensor_dim0[15:0]=256 (bits 63:48); atomic_barrier_addr=0
  g1[2] = (int)(S << 16);       // tensor_dim0[31:16]=0 | tensor_dim1[15:0]=2048
  g1[3] = (int)(KC << 16);      // tensor_dim1[31:16]=0 | tile_dim0=64
  g1[4] = 128;                  // tile_dim1=128 rows | tile_dim2=0
  g1[5] = RANK;                 // tensor_dim0_stride low32 = 256
  g1[6] = 0;                    // stride hi | tensor_dim1_stride lo (unused, 2D)
  g1[7] = 0;
  v4i z4 = {0, 0, 0, 0};        // groups 2/3: tensor is 2D
  v8i z8 = {0, 0, 0, 0, 0, 0, 0, 0};
  __builtin_amdgcn_tensor_load_to_lds(g0, g1, z4, z4, z8, /*cpol=*/0);
}

// ---------------- Kernel 2: TDM double-buffered f32 WMMA Gram + distance epilogue ----------------
__global__ __launch_bounds__(256)
void pairdist_wmma_kernel(const float* __restrict__ X,
                          const float* __restrict__ nrm,
                          float* __restrict__ out) {
  // 2 double-buffer slots x (A tile + B tile) x 128x68 f32 = 139264 B
  __shared__ float smem[2 * BUFF];

  const int b       = blockIdx.z;
  const int rowBase = blockIdx.y * TM;
  const int colBase = blockIdx.x * TN;
  const float* Xb   = X + (size_t)b * S * RANK;

  const int lane = threadIdx.x & 31;
  const int wid  = threadIdx.x >> 5;
  const int wr   = wid >> 2;              // 0..1 : 64 output rows per wave
  const int wc   = wid & 3;               // 0..3 : 32 output cols per wave
  const int lrow = lane & 15;
  const int lk   = (lane >> 4) << 1;

  const unsigned ldsBase = (unsigned)(uintptr_t)(&smem[0]);

  v8f acc[4][2] = {};

  int aOff[4], bOff[2];
  #pragma unroll
  for (int tm = 0; tm < 4; ++tm) aOff[tm] = (wr * 64 + tm * 16 + lrow) * LDA + lk;
  #pragma unroll
  for (int tn = 0; tn < 2; ++tn) bOff[tn] = (wc * 32 + tn * 16 + lrow) * LDA + lk;

  // Prologue: DMA chunk 0 into buffer 0 (EXEC-independent, issue from wave 0 only)
  if (wid == 0) {
    tdm_load_tile(Xb + (size_t)rowBase * RANK, ldsBase);
    tdm_load_tile(Xb + (size_t)colBase * RANK, ldsBase + TM * LDA * 4);
  }

  for (int c = 0; c < RANK / KC; ++c) {
    __syncthreads();   // all waves done reading buf[(c+1)&1] (used by chunk c-1)
    if (wid == 0) {
      if (c + 1 < RANK / KC) {
        const int kn = (c + 1) * KC;
        const unsigned nb0 = ldsBase + ((unsigned)((c + 1) & 1)) * (BUFF * 4);
        tdm_load_tile(Xb + (size_t)rowBase * RANK + kn, nb0);
        tdm_load_tile(Xb + (size_t)colBase * RANK + kn, nb0 + TM * LDA * 4);
        __builtin_amdgcn_s_wait_tensorcnt(2);   // chunk c done (in-order); c+1 in flight
      } else {
        __builtin_amdgcn_s_wait_tensorcnt(0);
      }
    }
    __syncthreads();   // publish buf[c&1]
    const float* As = smem + (c & 1) * BUFF;
    const float* Bs = As + TM * LDA;

    #pragma unroll
    for (int k = 0; k < KC; k += 4) {
      v2f af[4], bf[2];
      #pragma unroll
      for (int tm = 0; tm < 4; ++tm) af[tm] = *(const v2f*)(As + aOff[tm] + k);
      #pragma unroll
      for (int tn = 0; tn < 2; ++tn) bf[tn] = *(const v2f*)(Bs + bOff[tn] + k);
      #pragma unroll
      for (int tm = 0; tm < 4; ++tm)
        #pragma unroll
        for (int tn = 0; tn < 2; ++tn)
          acc[tm][tn] = __builtin_amdgcn_wmma_f32_16x16x4_f32(
              false, af[tm], false, bf[tn],
              (short)0, acc[tm][tn], false, false);
    }
  }

  // ---- Epilogue: stage 128x128 tile in LDS, stream 512B rows with NT stores ----
  __syncthreads();
  #pragma unroll
  for (int tm = 0; tm < 4; ++tm)
    #pragma unroll
    for (int tn = 0; tn < 2; ++tn)
      #pragma unroll
      for (int v = 0; v < 8; ++v) {
        int r = wr * 64 + tm * 16 + v + ((lane >> 4) << 3);
        int cc = wc * 32 + tn * 16 + lrow;
        smem[r * LDC + cc] = acc[tm][tn][v];
      }
  __syncthreads();

  const float* nbp = nrm + b * S;
  float* outB = out + ((size_t)b * S + rowBase) * S + colBase;
  #pragma unroll
  for (int i = 0; i < 16; ++i) {
    int idx = threadIdx.x + i * 256;   // one full 128-float row per wave per iter
    int r   = idx >> 5;
    int cc  = (idx & 31) << 2;
    v4f g  = *(const v4f*)(smem + r * LDC + cc);
    float ni = nbp[rowBase + r];
    v4f nj = *(const v4f*)(nbp + colBase + cc);
    v4f t  = (ni + nj) - 2.0f * g;
    v4f d;
    d.x = fmaxf(t.x, 0.f);
    d.y = fmaxf(t.y, 0.f);
    d.z = fmaxf(t.z, 0.f);
    d.w = fmaxf(t.w, 0.f);
    __builtin_nontemporal_store(d, (v4f*)(outB + (size_t)r * S + cc));
  }
}

extern "C" void kernel_launch(void* const* d_in, const int* in_sizes, int n_in,
                              void* d_out, int out_size, void* d_ws, size_t ws_size,
                              hipStream_t stream) {
  const float* X = (const float*)d_in[0];   // [8, 2048, 256] fp32
  float* out     = (float*)d_out;           // [8, 2048, 2048] fp32
  float* nrm     = (float*)d_ws;            // [8*2048] fp32 scratch (64 KB)

  row_sqnorm_kernel<<<dim3((NB * S) / 8), dim3(256), 0, stream>>>(X, nrm);
  pairdist_wmma_kernel<<<dim3(S / TN, S / TM, NB), dim3(256), 0, stream>>>(X, nrm, out);
}